// Spring_Mass_14817637171608
// MI455X (gfx1250) — compile-verified
//
#include <hip/hip_runtime.h>
#include <math.h>

#define NP      50000
#define KNB     16
#define NSTEP   32
#define BLOCK   256

__device__ __constant__ float kDT   = 1.0f / 60.0f;
#define EPS_F   1e-14f
#define EDGE_F  1e-6f

struct __align__(4) f3 { float x, y, z; };

// One simulation step. x_prev -> x_out, v updated in place (only own lane touches v[i]).
__global__ __launch_bounds__(BLOCK)
void Spring_step_kernel(const float* __restrict__ x_prev,   // [NP*3]
                        const int*   __restrict__ knn,      // [NP*KNB]
                        const float* __restrict__ L0,       // [NP*KNB]
                        const float* __restrict__ kspr,     // [NP*KNB]
                        const float* __restrict__ invm,     // [NP]
                        float*       __restrict__ v,        // [NP*3] in/out
                        float*       __restrict__ x_out,    // [NP*3]
                        const float* __restrict__ rebound_k,// [1]
                        const float* __restrict__ fric_k,   // [1]
                        const float* __restrict__ gptr)     // [1]
{
    int i = blockIdx.x * BLOCK + threadIdx.x;
    if (i >= NP) return;

    const f3 xi = ((const f3*)x_prev)[i];

    // Batch all per-thread contiguous loads upfront (vectorizes to global_load_b128)
    // so every memory op is in flight before the dependent sqrt/div chain starts.
    int   idx[KNB];
    float l0r[KNB];
    float ksr[KNB];
#pragma unroll
    for (int k = 0; k < KNB; ++k) idx[k] = knn[i * KNB + k];
#pragma unroll
    for (int k = 0; k < KNB; ++k) l0r[k] = L0[i * KNB + k];
#pragma unroll
    for (int k = 0; k < KNB; ++k) ksr[k] = kspr[i * KNB + k];

    // gfx1250 path: warm the gather cache lines (global_prefetch_b8).
#pragma unroll
    for (int k = 0; k < KNB; ++k)
        __builtin_prefetch(&x_prev[3 * idx[k]], 0, 3);

    float fx = 0.0f, fy = 0.0f, fz = 0.0f;
#pragma unroll
    for (int k = 0; k < KNB; ++k) {
        const f3 nb = ((const f3*)x_prev)[idx[k]];           // global_load_b96 gather
        float dx = nb.x - xi.x;
        float dy = nb.y - xi.y;
        float dz = nb.z - xi.z;
        float len = sqrtf(dx * dx + dy * dy + dz * dz + EPS_F);
        float dl  = len - l0r[k];
        float sgn = (dl > 0.0f) ? 1.0f : ((dl < 0.0f) ? -1.0f : 0.0f);
        float fm  = sgn * sqrtf(fabsf(dl) + EPS_F);          // POWER = 0.5
        float w   = ksr[k] * fm;
        fx += w * (dx / len);
        fy += w * (dy / len);
        fz += w * (dz / len);
    }

    const float im = invm[i];
    const float g  = gptr[0];
    float ax = fx * im;
    float ay = fy * im - g;          // g_acc = -g * (0,1,0)
    float az = fz * im;

    const f3 vi = ((const f3*)v)[i];
    float vx = vi.x + ax * kDT;
    float vy = vi.y + ay * kDT;
    float vz = vi.z + az * kDT;

    float nx = xi.x + vx * kDT;
    float ny = xi.y + vy * kDT;
    float nz = xi.z + vz * kDT;

    const bool below = (ny < 0.0f);
    const float rest = 1.0f / (1.0f + expf(-rebound_k[0]));  // sigmoid
    const float fric = 1.0f / (1.0f + expf(-fric_k[0]));     // sigmoid

    vy = below ? (-rest * vy) : vy;
    vx = below ? (fric * vx)  : vx;
    vz = below ? (fric * vz)  : vz;
    ny = below ? EDGE_F       : ny;

    f3 vo; vo.x = vx; vo.y = vy; vo.z = vz;
    ((f3*)v)[i] = vo;

    f3 xo; xo.x = nx; xo.y = ny; xo.z = nz;
    ((f3*)x_out)[i] = xo;
}

// One-time (per launch) precompute: spring constants, inverse masses, initial velocity.
__global__ __launch_bounds__(BLOCK)
void Spring_prep_kernel(const float* __restrict__ gk,   // [NP*KNB] log10 stiffness
                        const float* __restrict__ gm,   // [NP]     log10 mass
                        const float* __restrict__ iv,   // [3]      init velocity
                        float* __restrict__ kspr,       // [NP*KNB]
                        float* __restrict__ invm,       // [NP]
                        float* __restrict__ v)          // [NP*3]
{
    int i = blockIdx.x * BLOCK + threadIdx.x;
    if (i >= NP) return;
    invm[i] = 1.0f / powf(10.0f, gm[i]);
    f3 vo; vo.x = iv[0]; vo.y = iv[1]; vo.z = iv[2];
    ((f3*)v)[i] = vo;
#pragma unroll
    for (int k = 0; k < KNB; ++k)
        kspr[i * KNB + k] = powf(10.0f, gk[i * KNB + k]);
}

extern "C" void kernel_launch(void* const* d_in, const int* in_sizes, int n_in,
                              void* d_out, int out_size, void* d_ws, size_t ws_size,
                              hipStream_t stream) {
    // setup_inputs() order:
    const float* init_xyz   = (const float*)d_in[0];   // [NP,3]
    const float* init_vel   = (const float*)d_in[1];   // [3]
    const float* origin_len = (const float*)d_in[2];   // [NP,KNB]
    const float* global_k   = (const float*)d_in[3];   // [NP,KNB]
    const float* global_m   = (const float*)d_in[4];   // [NP]
    const float* rebound_k  = (const float*)d_in[5];   // [1]
    const float* fric_k     = (const float*)d_in[6];   // [1]
    const float* g          = (const float*)d_in[7];   // [1]
    const int*   knn_index  = (const int*)d_in[8];     // [NP,KNB]
    // d_in[9] = n_step (compile-time NSTEP=32)

    float* out = (float*)d_out;                        // [NSTEP, NP, 3]

    // Workspace layout (256B-aligned): v | kspr | invm   (~4 MB total)
    char* ws = (char*)d_ws;
    size_t off = 0;
    float* v    = (float*)(ws + off); off += ((size_t)NP * 3 * sizeof(float) + 255) & ~(size_t)255;
    float* kspr = (float*)(ws + off); off += ((size_t)NP * KNB * sizeof(float) + 255) & ~(size_t)255;
    float* invm = (float*)(ws + off);

    dim3 blk(BLOCK);
    dim3 grd((NP + BLOCK - 1) / BLOCK);

    Spring_prep_kernel<<<grd, blk, 0, stream>>>(global_k, global_m, init_vel,
                                                kspr, invm, v);

    const float* x_prev = init_xyz;
    for (int s = 0; s < NSTEP; ++s) {
        float* x_out = out + (size_t)s * NP * 3;
        Spring_step_kernel<<<grd, blk, 0, stream>>>(x_prev, knn_index, origin_len,
                                                    kspr, invm, v, x_out,
                                                    rebound_k, fric_k, g);
        x_prev = x_out;   // d_out doubles as the position ping-pong chain
    }
}